// TravelTimeDD_25331717111921
// MI455X (gfx1250) — compile-verified
//
#include <hip/hip_runtime.h>

// Eikonal grid constants (match reference)
#define TT_NR      1000
#define TT_NZ      100
#define TT_H       0.3f

#define N_STATION  500
#define BLOCK      256

// Bilinear interpolation of table[p] at (r, z); table layout [2, NR, NZ] row-major.
__device__ __forceinline__ float bilinear_tt(const float* __restrict__ table,
                                             int p, float r, float z) {
    float x = r / TT_H;                // RGRID0 == 0
    float y = z / TT_H;                // ZGRID0 == 0
    int ix = (int)floorf(x);
    int iy = (int)floorf(y);
    ix = min(max(ix, 0), TT_NR - 2);
    iy = min(max(iy, 0), TT_NZ - 2);
    float dx = fminf(fmaxf(x - (float)ix, 0.0f), 1.0f);
    float dy = fminf(fmaxf(y - (float)iy, 0.0f), 1.0f);
    const float* base = table + ((size_t)p * TT_NR + (size_t)ix) * TT_NZ + (size_t)iy;
    float v00 = base[0];
    float v01 = base[1];
    float v10 = base[TT_NZ];
    float v11 = base[TT_NZ + 1];
    float vx0 = fmaf(dx, v10 - v00, v00);   // interp along r at iy
    float vx1 = fmaf(dx, v11 - v01, v01);   // interp along r at iy+1
    return fmaf(dy, vx1 - vx0, vx0);
}

__global__ __launch_bounds__(BLOCK)
void tt_dd_main(const int*   __restrict__ station_index,   // [N]
                const int*   __restrict__ event_index,     // [N,2]
                const int*   __restrict__ phase_type,      // [N]
                const float* __restrict__ phase_time,      // [N]
                const float* __restrict__ phase_weight,    // [N]
                const float* __restrict__ event_loc,       // [N_EVENT,3]
                const float* __restrict__ event_time,      // [N_EVENT,1]
                const float* __restrict__ station_loc,     // [N_STATION,3]
                const float* __restrict__ station_dt,      // [N_STATION,1]
                const float* __restrict__ tt_table,        // [2,NR,NZ]
                float*       __restrict__ pred,            // [N]
                float*       __restrict__ partial,         // [numBlocks]
                int n) {
    // LDS-resident station tables (8 KB) + reduction scratch.
    __shared__ __align__(16) float s_stloc[3 * N_STATION];   // 1500 floats
    __shared__ __align__(16) float s_stdt[N_STATION];        //  500 floats
    __shared__ float sred[BLOCK];

    const int tid = threadIdx.x;

    // ---- Stage station tables into LDS with CDNA5 async global->LDS DMA ----
    // Low 32 bits of a flat LDS pointer are the DS byte offset.
    {
        unsigned loc_lds = (unsigned)(size_t)&s_stloc[0];
        unsigned dt_lds  = (unsigned)(size_t)&s_stdt[0];
        // 1500 floats = 375 x b128 chunks
        for (int c = tid; c < (3 * N_STATION) / 4; c += BLOCK) {
            unsigned long long ga = (unsigned long long)(size_t)(station_loc + 4 * c);
            unsigned la = loc_lds + 16u * (unsigned)c;
            asm volatile("global_load_async_to_lds_b128 %0, %1, off"
                         :: "v"(la), "v"(ga) : "memory");
        }
        // 500 floats = 125 x b128 chunks
        for (int c = tid; c < N_STATION / 4; c += BLOCK) {
            unsigned long long ga = (unsigned long long)(size_t)(station_dt + 4 * c);
            unsigned la = dt_lds + 16u * (unsigned)c;
            asm volatile("global_load_async_to_lds_b128 %0, %1, off"
                         :: "v"(la), "v"(ga) : "memory");
        }
    }

    int i = blockIdx.x * BLOCK + tid;
    float contrib = 0.0f;

    // Issue the coalesced per-pick loads while the async DMA is in flight.
    int   st = 0, p = 0, e0 = 0, e1 = 0;
    float ptime = 0.0f, pw = 0.0f;
    bool active = (i < n);
    if (active) {
        st    = station_index[i];
        p     = phase_type[i];
        e0    = event_index[2 * i + 0];
        e1    = event_index[2 * i + 1];
        ptime = phase_time[i];
        pw    = phase_weight[i];
    }

    // Wait for this wave's async DMA, then barrier so all waves see full LDS tables.
    asm volatile("s_wait_asynccnt 0x0" ::: "memory");
    __syncthreads();

    if (active) {
        float sx  = s_stloc[3 * st + 0];
        float sy  = s_stloc[3 * st + 1];
        float sz  = s_stloc[3 * st + 2];
        float sdt = s_stdt[st];

        int ev[2] = { e0, e1 };
        float t[2];
        #pragma unroll
        for (int k = 0; k < 2; ++k) {
            int e = ev[k];
            float ex = event_loc[3 * e + 0];
            float ey = event_loc[3 * e + 1];
            float ez = event_loc[3 * e + 2];
            float dxe = ex - sx;
            float dye = ey - sy;
            float r = sqrtf(fmaf(dxe, dxe, dye * dye));  // epicentral distance
            float z = ez - sz;
            float tti = bilinear_tt(tt_table, p, r, z);
            t[k] = event_time[e] + tti + sdt;
        }

        float predv = t[0] - t[1];       // double difference (sdt cancels exactly)
        pred[i] = predv;

        float diff = predv - ptime;
        float ad   = fabsf(diff);
        float hub  = (ad < 1.0f) ? (0.5f * diff * diff) : (ad - 0.5f);
        contrib = hub * pw;
    }

    // Block-level tree reduction in LDS (deterministic within block).
    sred[tid] = contrib;
    __syncthreads();
    #pragma unroll
    for (int off = BLOCK / 2; off > 0; off >>= 1) {
        if (tid < off) sred[tid] += sred[tid + off];
        __syncthreads();
    }
    if (tid == 0) partial[blockIdx.x] = sred[0];
}

// Deterministic final reduction: fixed strided ownership, then LDS tree.
__global__ __launch_bounds__(BLOCK)
void tt_dd_reduce(const float* __restrict__ partial, int nb,
                  float* __restrict__ loss_out) {
    __shared__ float sred[BLOCK];
    float acc = 0.0f;
    for (int i = threadIdx.x; i < nb; i += BLOCK)
        acc += partial[i];
    sred[threadIdx.x] = acc;
    __syncthreads();
    #pragma unroll
    for (int off = BLOCK / 2; off > 0; off >>= 1) {
        if (threadIdx.x < off) sred[threadIdx.x] += sred[threadIdx.x + off];
        __syncthreads();
    }
    if (threadIdx.x == 0) loss_out[0] = sred[0];
}

extern "C" void kernel_launch(void* const* d_in, const int* in_sizes, int n_in,
                              void* d_out, int out_size, void* d_ws, size_t ws_size,
                              hipStream_t stream) {
    const int*   station_index = (const int*)  d_in[0];
    const int*   event_index   = (const int*)  d_in[1];
    const int*   phase_type    = (const int*)  d_in[2];
    const float* phase_time    = (const float*)d_in[3];
    const float* phase_weight  = (const float*)d_in[4];
    const float* event_loc     = (const float*)d_in[5];
    const float* event_time    = (const float*)d_in[6];
    const float* station_loc   = (const float*)d_in[7];
    const float* station_dt    = (const float*)d_in[8];
    const float* tt_table      = (const float*)d_in[9];
    // d_in[10] = grad_r_table, d_in[11] = grad_z_table: unused by forward reference.

    int n  = in_sizes[0];               // N_PICK
    int nb = (n + BLOCK - 1) / BLOCK;

    float* pred    = (float*)d_out;     // [n]
    float* loss    = pred + n;          // scalar at d_out[n]
    float* partial = (float*)d_ws;      // [nb] floats (fully rewritten each call)

    tt_dd_main<<<nb, BLOCK, 0, stream>>>(station_index, event_index, phase_type,
                                         phase_time, phase_weight,
                                         event_loc, event_time,
                                         station_loc, station_dt,
                                         tt_table, pred, partial, n);
    tt_dd_reduce<<<1, BLOCK, 0, stream>>>(partial, nb, loss);
}